// Speaker2Dubber_3951369912711
// MI455X (gfx1250) — compile-verified
//
#include <hip/hip_runtime.h>
#include <math.h>

#define NEGV (-1e9f)

typedef float v2f __attribute__((ext_vector_type(2)));
typedef float v8f __attribute__((ext_vector_type(8)));
typedef unsigned v4u __attribute__((ext_vector_type(4)));
typedef int v4i __attribute__((ext_vector_type(4)));
typedef int v8i __attribute__((ext_vector_type(8)));

// ---------------------------------------------------------------------------
// K1: per-batch duration preprocessing.
//   cof = sum(d)/lip_len ; lt = round(d/cof) ; correct argmax row by dif ;
//   ends = cumsum(lt) ; begins = ends - lt.
// One block per batch, blockDim == Ts (512, power of 2).
// ---------------------------------------------------------------------------
__global__ void prep_kernel(const int* __restrict__ d_targets,
                            const int* __restrict__ lip_lens,
                            int* __restrict__ begins, int* __restrict__ ends,
                            float* __restrict__ cofs, int Ts) {
  const int b = blockIdx.x;
  const int s = threadIdx.x;
  __shared__ int   si[512];
  __shared__ float sf[512];
  __shared__ int   sidx[512];

  int d  = d_targets[b * Ts + s];
  int ll = lip_lens[b];

  // sum of d (int tree reduce)
  si[s] = d; __syncthreads();
  for (int off = Ts >> 1; off > 0; off >>= 1) {
    if (s < off) si[s] += si[s + off];
    __syncthreads();
  }
  int sum_d = si[0];
  __syncthreads();

  float cof = (float)sum_d / (float)ll;
  float ltf = rintf((float)d / cof);   // round-to-nearest-even == jnp.round

  // sum of ltf
  sf[s] = ltf; __syncthreads();
  for (int off = Ts >> 1; off > 0; off >>= 1) {
    if (s < off) sf[s] += sf[s + off];
    __syncthreads();
  }
  float S = sf[0];
  __syncthreads();

  // argmax with first-index tie-break
  sf[s] = ltf; sidx[s] = s; __syncthreads();
  for (int off = Ts >> 1; off > 0; off >>= 1) {
    if (s < off) {
      float a = sf[s], c = sf[s + off];
      int ia = sidx[s], ic = sidx[s + off];
      if (c > a || (c == a && ic < ia)) { sf[s] = c; sidx[s] = ic; }
    }
    __syncthreads();
  }
  int maxidx = sidx[0];
  __syncthreads();

  float dif = (float)ll - S;
  if (s == maxidx) ltf += dif;
  int lt = (int)ltf;

  // inclusive scan (Hillis-Steele)
  si[s] = lt; __syncthreads();
  for (int off = 1; off < Ts; off <<= 1) {
    int t = (s >= off) ? si[s - off] : 0;
    __syncthreads();
    si[s] += t;
    __syncthreads();
  }
  int e = si[s];
  ends[b * Ts + s]   = e;
  begins[b * Ts + s] = e - lt;
  if (s == 0) cofs[b] = cof;
}

// ---------------------------------------------------------------------------
// K1b: mean_cof = mean_b(cofs)
// ---------------------------------------------------------------------------
__global__ void meancof_kernel(const float* __restrict__ cofs,
                               float* __restrict__ mc, int B) {
  __shared__ float sf[64];
  int t = threadIdx.x;
  float acc = 0.f;
  for (int i = t; i < B; i += 64) acc += cofs[i];
  sf[t] = acc; __syncthreads();
  for (int off = 32; off > 0; off >>= 1) {
    if (t < off) sf[t] += sf[t + off];
    __syncthreads();
  }
  if (t == 0) mc[0] = sf[0] / (float)B;
}

// ---------------------------------------------------------------------------
// K2: row-norm reciprocals. One wave32 per row; rln = 1/||lip||,
// rtn = 1/(||phon||+1)  (applied at GEMM store: separable rank-1 scaling)
// ---------------------------------------------------------------------------
__global__ void rownorm_kernel(const float* __restrict__ lip,
                               const float* __restrict__ phon,
                               float* __restrict__ rln, float* __restrict__ rtn,
                               int nLip, int nText, int D) {
  int lane = threadIdx.x & 31;
  int wave = threadIdx.x >> 5;
  int row  = blockIdx.x * 8 + wave;
  const float* base;
  if (row < nLip)              base = lip  + (size_t)row * D;
  else if (row < nLip + nText) base = phon + (size_t)(row - nLip) * D;
  else return;
  float acc = 0.f;
  for (int k = lane; k < D; k += 32) { float x = base[k]; acc += x * x; }
  for (int off = 16; off > 0; off >>= 1) acc += __shfl_xor(acc, off, 32);
  if (lane == 0) {
    float n = sqrtf(acc);
    if (row < nLip) rln[row]        = 1.0f / n;
    else            rtn[row - nLip] = 1.0f / (n + 1.0f);
  }
}

// ---------------------------------------------------------------------------
// TDM descriptor builders (CDNA5 D#, CDNA5 ISA).
// 2D tile: tile_dim0 x tile_dim1 elements (4B), row stride = strideElems,
// LDS padding: 4 DWORDs every 32 DWORDs -> 36-DWORD LDS row pitch.
// amdgpu-toolchain builtin arity: 6 args (g0 v4u, g1 v8i, v4i, v4i, v8i, cpol)
// ---------------------------------------------------------------------------
#if __has_builtin(__builtin_amdgcn_tensor_load_to_lds)
#define USE_TDM 1
__device__ __forceinline__ v4u tdm_g0(unsigned ldsaddr, const void* gptr) {
  unsigned long long ga = (unsigned long long)(uintptr_t)gptr;
  v4u g;
  g[0] = 1u;                                   // count=1, user mode
  g[1] = ldsaddr;                              // LDS byte address
  g[2] = (unsigned)ga;                         // global_addr[31:0]
  g[3] = (unsigned)(ga >> 32) | (2u << 30);    // global_addr[56:32] | type=2
  return g;
}
__device__ __forceinline__ v8i tdm_g1(int tile1, int strideElems) {
  // data_size=2 (4B) | pad_enable | pad_interval=4 (32 DW) | pad_amount=3 (4 DW)
  const unsigned td0 = 1u << 30, td1 = 1u << 30;  // huge dims: no OOB clip
  v8i g;
  g[0] = (int)((2u << 16) | (1u << 20) | (4u << 22) | (3u << 25));
  g[1] = (int)((td0 & 0xffffu) << 16);                       // tensor_dim0 lo
  g[2] = (int)((td0 >> 16) | ((td1 & 0xffffu) << 16));       // dim0 hi | dim1 lo
  g[3] = (int)((td1 >> 16) | (32u << 16));                   // dim1 hi | tile_dim0=32
  g[4] = tile1;                                              // tile_dim1 | tile_dim2=0
  g[5] = strideElems;                                        // tensor_dim0_stride lo
  g[6] = 0;                                                  // stride hi | dim1_stride lo
  g[7] = 0;
  return g;
}
#else
#define USE_TDM 0
#endif

// ---------------------------------------------------------------------------
// K3: fused similarity GEMM (WMMA f32 16x16x4) + gt_similarity.
// Block: 256 threads = 8 waves; tile 64(M=s) x 128(N=l); K staged 32 in LDS.
// Tiles staged RAW and row-major (TDM tile copies with HW padding when
// available); normalization applied at store. Each wave: 32x32 output.
// LDS pitch 36 -> all fragment reads are aligned ds_load_b64, conflict-free.
// ---------------------------------------------------------------------------
__global__ __launch_bounds__(256)
void sim_wmma_kernel(const float* __restrict__ phon, const float* __restrict__ lip,
                     const float* __restrict__ rtn,  const float* __restrict__ rln,
                     const int* __restrict__ begins, const int* __restrict__ ends,
                     float* __restrict__ sim, float* __restrict__ gts,
                     int Ts, int Tl, int D) {
  __shared__ float As[64][36];    // [M][K] pitch 36
  __shared__ float Bl[128][36];   // [N][K] pitch 36 (row-major lip tile)
  const int b   = blockIdx.z;
  const int s0  = blockIdx.y * 64;
  const int l0  = blockIdx.x * 128;
  const int tid = threadIdx.x;
  const int lane = tid & 31;
  const int w  = tid >> 5;
  const int wm = w >> 2;          // 0..1
  const int wn = w & 3;           // 0..3
  const int hi = lane >> 4;       // 0/1 (half-wave selects K pair)
  const int lo = lane & 15;

  v8f acc[2][2];
  acc[0][0] = {}; acc[0][1] = {}; acc[1][0] = {}; acc[1][1] = {};

  const float* Abase = phon + ((size_t)b * Ts + s0) * D;
  const float* Bbase = lip  + ((size_t)b * Tl + l0) * D;

#if USE_TDM
  const unsigned asoff = (unsigned)(uintptr_t)(void*)As;
  const unsigned bloff = (unsigned)(uintptr_t)(void*)Bl;
  const v8i g1a = tdm_g1(64,  D);
  const v8i g1b = tdm_g1(128, D);
  const v4i z4 = {0, 0, 0, 0};
  const v8i z8 = {0, 0, 0, 0, 0, 0, 0, 0};
#endif

  for (int kt = 0; kt < D; kt += 32) {
#if USE_TDM
    if (w == 0) {
      __builtin_amdgcn_tensor_load_to_lds(tdm_g0(asoff, Abase + kt), g1a, z4, z4, z8, 0);
    } else if (w == 1) {
      __builtin_amdgcn_tensor_load_to_lds(tdm_g0(bloff, Bbase + kt), g1b, z4, z4, z8, 0);
    }
    __builtin_amdgcn_s_wait_tensorcnt(0);
    __syncthreads();
#else
    // manual fallback: coalesced row-major tile copies
    for (int i = 0; i < 8; ++i) {               // A: 64x32
      int idx = i * 256 + tid;
      int r = idx >> 5, c = idx & 31;
      As[r][c] = Abase[(size_t)r * D + kt + c];
    }
    for (int i = 0; i < 16; ++i) {              // B: 128x32
      int idx = i * 256 + tid;
      int n = idx >> 5, k = idx & 31;
      Bl[n][k] = Bbase[(size_t)n * D + kt + k];
    }
    __syncthreads();
#endif
    for (int kk = 0; kk < 32; kk += 4) {
      int k0 = kk + hi * 2;       // even -> 8B-aligned ds_load_b64
      int m0 = wm * 32 + lo;
      int n0 = wn * 32 + lo;
      v2f a0 = *(const v2f*)&As[m0][k0];
      v2f a1 = *(const v2f*)&As[m0 + 16][k0];
      v2f b0 = *(const v2f*)&Bl[n0][k0];        // B[k][n] = lip[n][k]
      v2f b1 = *(const v2f*)&Bl[n0 + 16][k0];
      acc[0][0] = __builtin_amdgcn_wmma_f32_16x16x4_f32(false, a0, false, b0, (short)0, acc[0][0], false, false);
      acc[0][1] = __builtin_amdgcn_wmma_f32_16x16x4_f32(false, a0, false, b1, (short)0, acc[0][1], false, false);
      acc[1][0] = __builtin_amdgcn_wmma_f32_16x16x4_f32(false, a1, false, b0, (short)0, acc[1][0], false, false);
      acc[1][1] = __builtin_amdgcn_wmma_f32_16x16x4_f32(false, a1, false, b1, (short)0, acc[1][1], false, false);
    }
    __syncthreads();
  }

  // C/D layout: VGPR r -> row r + 8*hi ; col = lo. Normalize at store.
  for (int jn = 0; jn < 2; ++jn) {
    int col = l0 + wn * 32 + jn * 16 + lo;
    float rl = rln[(size_t)b * Tl + col];
    for (int im = 0; im < 2; ++im)
      for (int r = 0; r < 8; ++r) {
        int row = s0 + wm * 32 + im * 16 + r + 8 * hi;
        float rt = rtn[(size_t)b * Ts + row];
        size_t o = ((size_t)b * Ts + row) * Tl + col;
        sim[o] = acc[im][jn][r] * rt * rl;
        int bg = begins[b * Ts + row];
        int en = ends[b * Ts + row];
        gts[o] = (col >= bg && col < en) ? 1.0f : 0.0f;
      }
  }
}

// ---------------------------------------------------------------------------
// K4: zero-fill the alignment output (float4 stores)
// ---------------------------------------------------------------------------
__global__ void zero_kernel(float4* __restrict__ p, size_t n4) {
  size_t i = (size_t)blockIdx.x * blockDim.x + threadIdx.x;
  size_t stride = (size_t)gridDim.x * blockDim.x;
  float4 z = {0.f, 0.f, 0.f, 0.f};
  for (; i < n4; i += stride) p[i] = z;
}

// ---------------------------------------------------------------------------
// K5: monotonic-alignment forward DP. One block per batch, thread = x.
// v kept in registers, neighbor via __shfl_up (wave32), wave boundaries via
// tiny double-buffered LDS, one barrier per j step. Similarity staged in LDS
// 16 columns at a time (coalesced, prefetched). Directions via ballot_w32.
// ---------------------------------------------------------------------------
__global__ __launch_bounds__(512)
void forward_dp_kernel(const float* __restrict__ sim,
                       const int* __restrict__ src_lens,
                       const int* __restrict__ lip_lens,
                       unsigned* __restrict__ dirs, int Ts, int Tl) {
  constexpr int CH = 16;
  __shared__ float vals[512][CH + 1];
  __shared__ float bbuf[2][32];
  const int b = blockIdx.x;
  const int x = threadIdx.x;          // blockDim.x == Ts
  const int lane = x & 31, w = x >> 5;
  const int NW = Ts >> 5;
  const int sl = src_lens[b], ll = lip_lens[b];
  const bool vx = x < sl;

  if (x < 64) bbuf[x >> 5][x & 31] = 0.f;   // scan carry init = 0
  float v = 0.f;
  const float* simb = sim + (size_t)b * Ts * Tl;

  for (int j0 = 0; j0 < Tl; j0 += CH) {
    __syncthreads();
    for (int i = 0; i < CH; ++i) {
      int idx = i * Ts + x;
      int xr  = idx / CH;
      int cc  = idx % CH;
      vals[xr][cc] = simb[(size_t)xr * Tl + j0 + cc];
      if (j0 + CH < Tl)  // pull next chunk toward L2/L0 (global_prefetch_b8)
        __builtin_prefetch(&simb[(size_t)xr * Tl + j0 + CH + cc], 0, 1);
    }
    __syncthreads();
    for (int cc = 0; cc < CH; ++cc) {
      int j = j0 + cc;
      float vprev = __shfl_up(v, 1, 32);
      if (lane == 0) vprev = (w == 0) ? NEGV : bbuf[(j + 1) & 1][w - 1];
      bool stay  = v >= vprev;
      float vmax = stay ? v : vprev;
      bool valid = vx && (j < ll);
      float val  = valid ? vals[x][cc] : 0.f;   // value * mask
      float vnew = (x <= j) ? (vmax + val) : NEGV;
      unsigned bm = __builtin_amdgcn_ballot_w32(stay || !valid); // forced stay when masked
      if (lane == 0)  dirs[((size_t)b * Tl + j) * NW + w] = bm;
      if (lane == 31) bbuf[j & 1][w] = vnew;
      v = vnew;
      __syncthreads();
    }
  }
}

// ---------------------------------------------------------------------------
// K6: backward path walk + d_predictions. One block per batch; packed dirs
// staged into LDS in 256-row chunks so dependent loads hit LDS.
// ---------------------------------------------------------------------------
__global__ __launch_bounds__(256)
void backward_kernel(const unsigned* __restrict__ dirs,
                     const int* __restrict__ src_lens,
                     const int* __restrict__ lip_lens,
                     const float* __restrict__ mc,
                     float* __restrict__ align_out, float* __restrict__ dpred,
                     int Ts, int Tl) {
  constexpr int CJ = 256;
  __shared__ unsigned sdirs[CJ * 16];
  __shared__ int cnt[512];
  const int b = blockIdx.x;
  const int tid = threadIdx.x;
  const int NW = Ts >> 5;
  const int sl = src_lens[b], ll = lip_lens[b];
  for (int s = tid; s < Ts; s += 256) cnt[s] = 0;

  int idx = sl - 1;
  const unsigned* db = dirs + (size_t)b * Tl * NW;
  float* ab = align_out + (size_t)b * Ts * Tl;

  for (int j0 = Tl - CJ; j0 >= 0; j0 -= CJ) {
    __syncthreads();
    for (int i = tid; i < CJ * NW; i += 256)
      sdirs[i] = db[(size_t)j0 * NW + i];
    __syncthreads();
    if (tid == 0) {
      for (int jj = CJ - 1; jj >= 0; --jj) {
        int j = j0 + jj;
        if (idx >= 0) {
          if (j < ll) {                 // path * mask
            ab[(size_t)idx * Tl + j] = 1.0f;
            cnt[idx]++;
          }
          unsigned word = sdirs[jj * NW + (idx >> 5)];
          int d = (word >> (idx & 31)) & 1;
          idx += d - 1;                 // stay(1) -> idx ; move(0) -> idx-1
        }
      }
    }
  }
  __syncthreads();
  float m = mc[0];
  for (int s = tid; s < Ts; s += 256)
    dpred[b * Ts + s] = (float)cnt[s] * m;
}

// ---------------------------------------------------------------------------
extern "C" void kernel_launch(void* const* d_in, const int* in_sizes, int n_in,
                              void* d_out, int out_size, void* d_ws, size_t ws_size,
                              hipStream_t stream) {
  const float* lip       = (const float*)d_in[0];
  const float* phon      = (const float*)d_in[1];
  const int*   d_targets = (const int*)d_in[2];
  const int*   src_lens  = (const int*)d_in[3];
  const int*   lip_lens  = (const int*)d_in[4];

  const int B  = in_sizes[3];
  const int Ts = in_sizes[2] / B;
  const int D  = in_sizes[1] / (B * Ts);
  const int Tl = in_sizes[0] / (B * D);

  float* out = (float*)d_out;
  size_t N1 = (size_t)B * Ts * Tl;
  float* sim = out;
  float* gts = out + N1;
  float* alg = out + 2 * N1;
  float* dpr = out + 3 * N1;

  // workspace layout (total ~4.6 MB for B=32,Ts=512,Tl=2048)
  char* ws = (char*)d_ws;
  auto alignup = [](size_t v) { return (v + 255) & ~(size_t)255; };
  size_t o = 0;
  float* rln    = (float*)(ws + o); o = alignup(o + (size_t)B * Tl * 4);
  float* rtn    = (float*)(ws + o); o = alignup(o + (size_t)B * Ts * 4);
  int*   begins = (int*)  (ws + o); o = alignup(o + (size_t)B * Ts * 4);
  int*   endsp  = (int*)  (ws + o); o = alignup(o + (size_t)B * Ts * 4);
  float* cofs   = (float*)(ws + o); o = alignup(o + (size_t)B * 4);
  float* mc     = (float*)(ws + o); o = alignup(o + 4);
  unsigned* dirs = (unsigned*)(ws + o);

  prep_kernel<<<B, Ts, 0, stream>>>(d_targets, lip_lens, begins, endsp, cofs, Ts);
  meancof_kernel<<<1, 64, 0, stream>>>(cofs, mc, B);

  int rows = B * (Tl + Ts);
  rownorm_kernel<<<(rows + 7) / 8, 256, 0, stream>>>(lip, phon, rln, rtn,
                                                     B * Tl, B * Ts, D);

  dim3 g(Tl / 128, Ts / 64, B);
  sim_wmma_kernel<<<g, 256, 0, stream>>>(phon, lip, rtn, rln, begins, endsp,
                                         sim, gts, Ts, Tl, D);

  zero_kernel<<<4096, 256, 0, stream>>>((float4*)alg, N1 / 4);

  forward_dp_kernel<<<B, Ts, 0, stream>>>(sim, src_lens, lip_lens, dirs, Ts, Tl);

  backward_kernel<<<B, 256, 0, stream>>>(dirs, src_lens, lip_lens, mc,
                                         alg, dpr, Ts, Tl);
}